// GCN3Layer_PyG_78219944394962
// MI455X (gfx1250) — compile-verified
//
#include <hip/hip_runtime.h>

typedef __attribute__((ext_vector_type(2))) float v2f;
typedef __attribute__((ext_vector_type(8))) float v8f;

#define NFEAT 512
#define NHID  128
#define NCLASS 10

// ---------------- prep kernels ----------------

// Wt[n*K + k] = W[k*Nc + n]
__global__ void transpose_w(const float* __restrict__ W, float* __restrict__ Wt,
                            int K, int Nc) {
  int idx = blockIdx.x * blockDim.x + threadIdx.x;
  if (idx < K * Nc) {
    int k = idx / Nc, n = idx % Nc;
    Wt[(size_t)n * K + k] = W[idx];
  }
}

// deg starts at 1.0 (self loop)
__global__ void deg_init(float* __restrict__ deg, int n) {
  int i = blockIdx.x * blockDim.x + threadIdx.x;
  if (i < n) deg[i] = 1.0f;
}

__global__ void deg_edges(const int* __restrict__ cols, float* __restrict__ deg, int E) {
  int e = blockIdx.x * blockDim.x + threadIdx.x;
  if (e < E) atomicAdd(&deg[cols[e]], 1.0f);
}

// in-place: deg -> d^{-1/2}  (deg >= 1 always, so no zero-guard needed)
__global__ void deg_fin(float* __restrict__ deg, int n) {
  int i = blockIdx.x * blockDim.x + threadIdx.x;
  if (i < n) deg[i] = rsqrtf(deg[i]);
}

// ---------------- WMMA GEMM: H[M x 128] = X[M x K] * Wt[128 x K]^T ----------------
// One wave computes a 16 x 128 output strip; fp32 tensor op V_WMMA_F32_16X16X4_F32.
// Requires M % 16 == 0 (true here: 50000 = 3125*16). Guard is wave-uniform so
// EXEC stays all-ones for every executed WMMA.
__global__ void __launch_bounds__(128) gemm_wmma_f32(const float* __restrict__ X,
                                                     const float* __restrict__ Wt,
                                                     float* __restrict__ H,
                                                     int M, int K) {
  int wave = (int)((blockIdx.x * (size_t)blockDim.x + threadIdx.x) >> 5);
  int lane = threadIdx.x & 31;
  if (wave * 16 >= M) return;

  int nlo = lane & 15;            // A: row M = lane&15 ; B: col N = lane&15
  int kh  = (lane >> 4) * 2;      // lanes 16-31 hold K = +2,+3 of each 4-chunk

  const float* xrow = X + (size_t)(wave * 16 + nlo) * K;

  v8f acc[8] = {};
  for (int k0 = 0; k0 < K; k0 += 4) {
    v2f a = *(const v2f*)(xrow + k0 + kh);
#pragma unroll
    for (int t = 0; t < 8; ++t) {
      v2f b = *(const v2f*)(Wt + (size_t)(t * 16 + nlo) * K + k0 + kh);
      acc[t] = __builtin_amdgcn_wmma_f32_16x16x4_f32(
          /*neg_a=*/false, a, /*neg_b=*/false, b,
          /*c_mod=*/(short)0, acc[t], /*reuse_a=*/false, /*reuse_b=*/false);
    }
  }

  // C/D layout: element e, lanes 0-15 -> M=e, lanes 16-31 -> M=e+8; N = lane&15
  int mbase = wave * 16 + ((lane >> 4) << 3);
#pragma unroll
  for (int t = 0; t < 8; ++t) {
#pragma unroll
    for (int e = 0; e < 8; ++e) {
      H[(size_t)(mbase + e) * NHID + t * 16 + nlo] = acc[t][e];
    }
  }
}

// ---------------- aggregation ----------------

// self-loop term: Y[i][f] = H[i][f] * dinv[i]^2
__global__ void agg_init(const float* __restrict__ H, const float* __restrict__ dinv,
                         float* __restrict__ Y, int total) {
  int idx = blockIdx.x * blockDim.x + threadIdx.x;
  if (idx < total) {
    int i = idx >> 7;  // /NHID
    float d = dinv[i];
    Y[idx] = H[idx] * d * d;
  }
}

// one wave per edge: Y[col] += H[row] * dinv[row]*dinv[col]
__global__ void __launch_bounds__(256) agg_edges(const int* __restrict__ rows,
                                                 const int* __restrict__ cols,
                                                 const float* __restrict__ dinv,
                                                 const float* __restrict__ H,
                                                 float* __restrict__ Y, int E) {
  int e = (int)((blockIdx.x * (size_t)blockDim.x + threadIdx.x) >> 5);
  int lane = threadIdx.x & 31;
  if (e >= E) return;
  int r = rows[e];
  int c = cols[e];
  float w = dinv[r] * dinv[c];
  const float4 h = *(const float4*)(H + (size_t)r * NHID + lane * 4);
  float* yp = Y + (size_t)c * NHID + lane * 4;
  atomicAdd(yp + 0, h.x * w);
  atomicAdd(yp + 1, h.y * w);
  atomicAdd(yp + 2, h.z * w);
  atomicAdd(yp + 3, h.w * w);
}

__global__ void agg_bias_act(float* __restrict__ Y, const float* __restrict__ b,
                             int total, int relu) {
  int idx = blockIdx.x * blockDim.x + threadIdx.x;
  if (idx < total) {
    float v = Y[idx] + b[idx & (NHID - 1)];
    Y[idx] = relu ? fmaxf(v, 0.0f) : v;
  }
}

// ---------------- final linear + log_softmax (one wave per node) ----------------
__global__ void __launch_bounds__(256) final_logits(const float* __restrict__ X,
                                                    const float* __restrict__ Wlt,
                                                    const float* __restrict__ bl,
                                                    float* __restrict__ out, int n) {
  int node = (int)((blockIdx.x * (size_t)blockDim.x + threadIdx.x) >> 5);
  int lane = threadIdx.x & 31;
  if (node >= n) return;

  float4 xv = *(const float4*)(X + (size_t)node * NHID + lane * 4);
  float logit[NCLASS];
#pragma unroll
  for (int c = 0; c < NCLASS; ++c) {
    float4 wv = *(const float4*)(Wlt + (size_t)c * NHID + lane * 4);
    float p = xv.x * wv.x + xv.y * wv.y + xv.z * wv.z + xv.w * wv.w;
#pragma unroll
    for (int off = 16; off > 0; off >>= 1) p += __shfl_xor(p, off, 32);
    logit[c] = p + bl[c];
  }
  float m = logit[0];
#pragma unroll
  for (int c = 1; c < NCLASS; ++c) m = fmaxf(m, logit[c]);
  float s = 0.0f;
#pragma unroll
  for (int c = 0; c < NCLASS; ++c) s += __expf(logit[c] - m);
  float lse = m + __logf(s);
  if (lane < NCLASS) out[(size_t)node * NCLASS + lane] = logit[lane] - lse;
}

// ---------------- launcher ----------------
extern "C" void kernel_launch(void* const* d_in, const int* in_sizes, int n_in,
                              void* d_out, int out_size, void* d_ws, size_t ws_size,
                              hipStream_t stream) {
  const float* x  = (const float*)d_in[0];
  const int*   ei = (const int*)d_in[1];  // edge_index flat: [row(E), col(E)]
  const float* W1 = (const float*)d_in[2];
  const float* b1 = (const float*)d_in[3];
  const float* W2 = (const float*)d_in[4];
  const float* b2 = (const float*)d_in[5];
  const float* W3 = (const float*)d_in[6];
  const float* b3 = (const float*)d_in[7];
  const float* Wl = (const float*)d_in[8];
  const float* bl = (const float*)d_in[9];
  float* out = (float*)d_out;

  const int E = in_sizes[1] / 2;
  const int N = in_sizes[0] / NFEAT;

  // carve scratch from d_ws (256B aligned slices), ~52 MB total
  char* ws = (char*)d_ws;
  size_t off = 0;
  auto alloc = [&](size_t bytes) -> float* {
    float* p = (float*)(ws + off);
    off += (bytes + 255) & ~(size_t)255;
    return p;
  };
  float* dinv = alloc((size_t)N * sizeof(float));
  float* Wt1  = alloc((size_t)NFEAT * NHID * sizeof(float));
  float* Wt2  = alloc((size_t)NHID * NHID * sizeof(float));
  float* Wt3  = alloc((size_t)NHID * NHID * sizeof(float));
  float* Wlt  = alloc((size_t)NHID * NCLASS * sizeof(float));
  float* H    = alloc((size_t)N * NHID * sizeof(float));  // GEMM output
  float* Y    = alloc((size_t)N * NHID * sizeof(float));  // aggregation output

  const int* rows = ei;
  const int* cols = ei + E;

  transpose_w<<<(NFEAT * NHID + 255) / 256, 256, 0, stream>>>(W1, Wt1, NFEAT, NHID);
  transpose_w<<<(NHID * NHID + 255) / 256, 256, 0, stream>>>(W2, Wt2, NHID, NHID);
  transpose_w<<<(NHID * NHID + 255) / 256, 256, 0, stream>>>(W3, Wt3, NHID, NHID);
  transpose_w<<<(NHID * NCLASS + 255) / 256, 256, 0, stream>>>(Wl, Wlt, NHID, NCLASS);

  deg_init<<<(N + 255) / 256, 256, 0, stream>>>(dinv, N);
  deg_edges<<<(E + 255) / 256, 256, 0, stream>>>(cols, dinv, E);
  deg_fin<<<(N + 255) / 256, 256, 0, stream>>>(dinv, N);

  const int tiles = (N + 15) / 16;
  const int gemm_grid = (tiles + 3) / 4;     // 4 waves (128 thr) per block
  const int tot = N * NHID;
  const int tgrid = (tot + 255) / 256;
  const int egrid = (int)(((size_t)E * 32 + 255) / 256);

  // layer 1: x[N,512] -> H -> Y (relu)
  gemm_wmma_f32<<<gemm_grid, 128, 0, stream>>>(x, Wt1, H, N, NFEAT);
  agg_init<<<tgrid, 256, 0, stream>>>(H, dinv, Y, tot);
  agg_edges<<<egrid, 256, 0, stream>>>(rows, cols, dinv, H, Y, E);
  agg_bias_act<<<tgrid, 256, 0, stream>>>(Y, b1, tot, 1);

  // layer 2: Y -> H -> Y (relu)
  gemm_wmma_f32<<<gemm_grid, 128, 0, stream>>>(Y, Wt2, H, N, NHID);
  agg_init<<<tgrid, 256, 0, stream>>>(H, dinv, Y, tot);
  agg_edges<<<egrid, 256, 0, stream>>>(rows, cols, dinv, H, Y, E);
  agg_bias_act<<<tgrid, 256, 0, stream>>>(Y, b2, tot, 1);

  // layer 3: Y -> H -> Y (no relu)
  gemm_wmma_f32<<<gemm_grid, 128, 0, stream>>>(Y, Wt3, H, N, NHID);
  agg_init<<<tgrid, 256, 0, stream>>>(H, dinv, Y, tot);
  agg_edges<<<egrid, 256, 0, stream>>>(rows, cols, dinv, H, Y, E);
  agg_bias_act<<<tgrid, 256, 0, stream>>>(Y, b3, tot, 0);

  // classifier + log_softmax
  final_logits<<<(int)(((size_t)N * 32 + 255) / 256), 256, 0, stream>>>(Y, Wlt, bl, out, N);
}